// ShadowQSP_19705309954073
// MI455X (gfx1250) — compile-verified
//
#include <hip/hip_runtime.h>
#include <cstdint>

// ShadowQSP: B=131072 independent 4x4 complex chains collapsed to an 8-real
// state recurrence. VALU-bound (~0.4 GFLOP f32, ~2.6 MB traffic -> memory is
// ~0.1us at 23.3 TB/s; packed-f32 compute dominates). CDNA5 paths: wave32
// one-element-per-lane, async global->LDS staging of step constants
// (ASYNCcnt + s_wait_asynccnt), LDS broadcast reads, packed-fp32 VOP3P
// (v_pk_fma_f32) fully-unrolled inner recurrence.

#define MAX_DEG 64

typedef float v2f __attribute__((ext_vector_type(2)));

// i*z for z=(re,im): (-im, re). Backend folds shuffle+fneg into VOP3P
// op_sel/neg source modifiers on the consuming v_pk op.
static __device__ __forceinline__ v2f jmul(v2f z) {
    v2f r; r.x = -z.y; r.y = z.x; return r;
}
static __device__ __forceinline__ v2f vfma(v2f a, v2f b, v2f c) {
    return __builtin_elementwise_fma(a, b, c);
}

template <int NPHI_CT>
__global__ __launch_bounds__(256) void ShadowQSP_kernel(
    const float* __restrict__ inp,        // [B,2] (theta, eps)
    const float* __restrict__ phis,       // [nphi]
    const float* __restrict__ init_ang,   // [B]
    float* __restrict__ out,              // complex64 as float pairs (or real)
    int B, int nphi, int cplx_out)
{
    __shared__ float s_phi[MAX_DEG];
    __shared__ v2f   s_cs[MAX_DEG];

    const int tid = (int)threadIdx.x;
    const int n   = (NPHI_CT > 0) ? NPHI_CT : nphi;

    // ---- Stage phis into LDS via the gfx1250 async global->LDS copy ----
#if defined(__HIP_DEVICE_COMPILE__) && defined(__gfx1250__)
    if (tid < n) {
        unsigned lds_addr = (unsigned)(uintptr_t)&s_phi[tid];
        unsigned long long gaddr = (unsigned long long)(uintptr_t)(phis + tid);
        asm volatile("global_load_async_to_lds_b32 %0, %1, off"
                     :: "v"(lds_addr), "v"(gaddr)
                     : "memory");
    }
    asm volatile("s_wait_asynccnt 0x0" ::: "memory");
#else
    if (tid < n) s_phi[tid] = phis[tid];
#endif
    __syncthreads();

    // cos/sin of each phi once per block (inputs in [0,pi]: native
    // v_sin_f32/v_cos_f32 fast path).
    if (tid < n) {
        float s, c;
        __sincosf(s_phi[tid], &s, &c);
        v2f cs; cs.x = c; cs.y = s;
        s_cs[tid] = cs;
    }
    __syncthreads();

    const int b = (int)blockIdx.x * (int)blockDim.x + tid;
    if (b >= B) return;

    const float2 te = ((const float2*)inp)[b];   // global_load_b64, coalesced
    float st, ct, se, ce, sa, ca;
    __sincosf(te.x, &st, &ct);                   // theta  in [0,pi]
    __sincosf(te.y, &se, &ce);                   // eps    in [0,pi]
    __sincosf(init_ang[b], &sa, &ca);            // in [0,pi/2]

    // cv = V(eps)*C(th) rows (pure-real / pure-imag entries):
    //   r0 = [ce, 0, i*A, -Bc]   r1 = [0, ce, -Bc, i*A]
    //   r2 = [i*se, 0, C2, i*D2] r3 = [0, i*se, i*D2, C2]
    const float A  = se * ct;
    const float Bc = se * st;
    const float C2 = ce * ct;
    const float D2 = ce * st;

    const v2f ceb  = {ce,  ce};
    const v2f seb  = {se,  se};
    const v2f Ab   = {A,   A};
    const v2f nBcb = {-Bc, -Bc};
    const v2f C2b  = {C2,  C2};
    const v2f D2b  = {D2,  D2};

    // psi0 = [cos a, 0, i sin a, 0], each component as (re, im)
    v2f p0 = {ca, 0.f};
    v2f p1 = {0.f, 0.f};
    v2f p2 = {0.f, sa};
    v2f p3 = {0.f, 0.f};

    auto step = [&](int k) {
        const v2f cs = s_cs[k];                  // ds broadcast (b128-merged)
        const v2f cb  = {cs.x, cs.x};
        const v2f sb  = {cs.y, cs.y};
        const v2f nsb = {-cs.y, -cs.y};

        // diag: p0,p2 *= e^{i phi} = c*z + s*(i z);  p1,p3 *= e^{-i phi}
        p0 = vfma(cb, p0, sb  * jmul(p0));
        p1 = vfma(cb, p1, nsb * jmul(p1));
        p2 = vfma(cb, p2, sb  * jmul(p2));
        p3 = vfma(cb, p3, nsb * jmul(p3));

        // psi <- cv * psi (structured: 12 packed FMAs)
        const v2f jp0 = jmul(p0), jp1 = jmul(p1), jp2 = jmul(p2), jp3 = jmul(p3);
        const v2f n0 = vfma(ceb, p0,  vfma(Ab,  jp2, nBcb * p3));
        const v2f n1 = vfma(ceb, p1,  vfma(Ab,  jp3, nBcb * p2));
        const v2f n2 = vfma(seb, jp0, vfma(C2b, p2,  D2b * jp3));
        const v2f n3 = vfma(seb, jp1, vfma(D2b, jp2, C2b * p3));
        p0 = n0; p1 = n1; p2 = n2; p3 = n3;
    };

    if constexpr (NPHI_CT > 0) {
        #pragma unroll
        for (int k = 0; k < NPHI_CT; ++k) step(k);   // fully unrolled
    } else {
        #pragma unroll 4
        for (int k = 0; k < nphi; ++k) step(k);
    }

    // <psi| diag(1,-1,1,-1) |psi>  (real by construction), packed reduce
    v2f e = p0 * p0;  e = vfma(p2, p2, e);
    v2f o = p1 * p1;  o = vfma(p3, p3, o);
    const float ex = (e.x - o.x) + (e.y - o.y);

    if (cplx_out) {
        ((float2*)out)[b] = make_float2(ex, 0.0f);  // global_store_b64
    } else {
        out[b] = ex;
    }
}

extern "C" void kernel_launch(void* const* d_in, const int* in_sizes, int n_in,
                              void* d_out, int out_size, void* d_ws, size_t ws_size,
                              hipStream_t stream) {
    (void)n_in; (void)d_ws; (void)ws_size;
    const float* inp        = (const float*)d_in[0];
    const float* phis       = (const float*)d_in[1];
    const float* init_ang   = (const float*)d_in[2];
    float* out              = (float*)d_out;

    const int B    = in_sizes[0] / 2;
    int nphi       = in_sizes[1];
    if (nphi > MAX_DEG) nphi = MAX_DEG;
    const int cplx = (out_size >= 2 * B) ? 1 : 0;

    const int threads = 256;
    const int blocks  = (B + threads - 1) / threads;
    if (nphi == MAX_DEG) {
        hipLaunchKernelGGL((ShadowQSP_kernel<MAX_DEG>), dim3(blocks), dim3(threads),
                           0, stream, inp, phis, init_ang, out, B, nphi, cplx);
    } else {
        hipLaunchKernelGGL((ShadowQSP_kernel<0>), dim3(blocks), dim3(threads),
                           0, stream, inp, phis, init_ang, out, B, nphi, cplx);
    }
}